// GraphActor_8143257993844
// MI455X (gfx1250) — compile-verified
//
#include <hip/hip_runtime.h>
#include <hip/hip_bf16.h>
#include <math.h>

typedef __attribute__((ext_vector_type(2))) float v2f;
typedef __attribute__((ext_vector_type(8))) float v8f;

#define HID 128
#define LOG_2PI 1.8378770664093453f

// ---------------- small utility kernels ----------------

__global__ void k_zero(float* __restrict__ p, int n) {
  int i = blockIdx.x * blockDim.x + threadIdx.x;
  if (i < n) p[i] = 0.0f;
}

__global__ void k_deg(const int* __restrict__ col, float* __restrict__ deg, int n_edges) {
  int e = blockIdx.x * blockDim.x + threadIdx.x;
  if (e < n_edges) atomicAdd(&deg[col[e]], 1.0f);
}

__global__ void k_isd(float* __restrict__ deg_isd, int n_nodes) {
  int i = blockIdx.x * blockDim.x + threadIdx.x;
  if (i < n_nodes) deg_isd[i] = rsqrtf(deg_isd[i] + 1.0f);  // self-loop degree
}

// ---------------- fp32 WMMA GEMM: Y[M,128] = (relu_x ? relu(X) : X) @ W[K,128] (+ bias) ----
// One wave computes a 16-row M-tile across all 8 column tiles (128 cols).
// W is staged in LDS pre-swizzled into B-fragment layout so each B-frag is a
// single conflict-free ds_load_b64 with an immediate offset:
//   frag f = kt*8 + nt, lane l:
//   sWv[f*32 + l] = { W[kt*4 + (l>>4)*2 + 0][nt*16 + (l&15)],
//                     W[kt*4 + (l>>4)*2 + 1][nt*16 + (l&15)] }
// A-frag (16x4 f32): lane l holds row m = l&15, K = k + (l>>4)*2 + {0,1}.
// C/D (16x16 f32):   lane l col n = l&15; VGPR r holds row m = r + 8*(l>>4).
__global__ void __launch_bounds__(256)
k_gemm_wmma(const float* __restrict__ X, const float* __restrict__ W,
            const float* __restrict__ bias, float* __restrict__ Y,
            int M, int K, int relu_x) {
  __shared__ v2f sWv[HID * HID / 2];  // 64 KB max; only (K/4)*8*32 frags used
  const int tid = threadIdx.x;
  const int nentries = (K >> 2) * 8 * 32;  // fragments * lanes
  for (int idx = tid; idx < nentries; idx += 256) {
    const int l = idx & 31;
    const int f = idx >> 5;
    const int kt = f >> 3, nt = f & 7;
    const int kb = (kt << 2) + ((l >> 4) << 1);
    const int n = (nt << 4) + (l & 15);
    v2f w;
    w.x = W[kb * HID + n];
    w.y = W[(kb + 1) * HID + n];
    sWv[idx] = w;
  }
  __syncthreads();

  const int wave = tid >> 5;
  const int lane = tid & 31;
  const int mtile = blockIdx.x * 8 + wave;
  const int mtiles = (M + 15) >> 4;
  if (mtile >= mtiles) return;  // whole-wave exit: EXEC stays all-ones for WMMA

  const int lmod = lane & 15;
  const int lhi  = lane >> 4;

  v8f acc[8] = {};

  int mrow = mtile * 16 + lmod;
  if (mrow >= M) mrow = M - 1;  // clamp (stores are predicated below)
  const float* xrow = X + (size_t)mrow * K;

  for (int k = 0, kt = 0; k < K; k += 4, ++kt) {
    v2f a = *(const v2f*)(xrow + k + (lhi << 1));
    if (relu_x) { a.x = fmaxf(a.x, 0.0f); a.y = fmaxf(a.y, 0.0f); }
    const v2f* bp = sWv + (kt << 8) + lane;  // + nt*32 below (imm offset nt*256B)
#pragma unroll
    for (int nt = 0; nt < 8; ++nt) {
      acc[nt] = __builtin_amdgcn_wmma_f32_16x16x4_f32(
          /*neg_a=*/false, a, /*neg_b=*/false, bp[nt << 5],
          /*c_mod=*/(short)0, acc[nt], /*reuse_a=*/false, /*reuse_b=*/false);
    }
  }

#pragma unroll
  for (int nt = 0; nt < 8; ++nt) {
    const int n = (nt << 4) + lmod;
    const float bv = bias ? bias[n] : 0.0f;
#pragma unroll
    for (int r = 0; r < 8; ++r) {
      const int m = mtile * 16 + r + (lhi << 3);
      if (m < M) Y[(size_t)m * HID + n] = acc[nt][r] + bv;
    }
  }
}

// ---------------- GCN aggregation ----------------

// agg[n][:] = hw[n][:] * isd[n]^2 + b[:]   (self-loop term + bias)
__global__ void __launch_bounds__(256)
k_agginit(const float* __restrict__ hw, const float* __restrict__ isd,
          const float* __restrict__ b, float* __restrict__ agg, int n_nodes) {
  int t = blockIdx.x * blockDim.x + threadIdx.x;  // over n_nodes*32
  if (t >= n_nodes * 32) return;
  const int node = t >> 5;
  const int c4 = (t & 31) << 2;
  float s = isd[node];
  s = s * s;
  float4 v = *(const float4*)(hw + (size_t)node * HID + c4);
  float4 bb = *(const float4*)(b + c4);
  float4 o;
  o.x = v.x * s + bb.x; o.y = v.y * s + bb.y;
  o.z = v.z * s + bb.z; o.w = v.w * s + bb.w;
  *(float4*)(agg + (size_t)node * HID + c4) = o;
}

// agg[col[e]][:] += hw[row[e]][:] * (isd[row]*isd[col]); 32 threads/edge, float4 each
__global__ void __launch_bounds__(256)
k_scatter(const float* __restrict__ hw, const int* __restrict__ row,
          const int* __restrict__ col, const float* __restrict__ isd,
          float* __restrict__ agg, int n_edges) {
  long long t = (long long)blockIdx.x * 256 + threadIdx.x;
  const int e = (int)(t >> 5);
  if (e >= n_edges) return;
  const int c4 = ((int)t & 31) << 2;
  const int r = row[e], c = col[e];
  const float coef = isd[r] * isd[c];
  float4 v = *(const float4*)(hw + (size_t)r * HID + c4);
  float* dst = agg + (size_t)c * HID + c4;
  atomicAdd(dst + 0, v.x * coef);
  atomicAdd(dst + 1, v.y * coef);
  atomicAdd(dst + 2, v.z * coef);
  atomicAdd(dst + 3, v.w * coef);
}

// ---------------- sampling head ----------------
// z = relu(h[map[i]]); out[i] = tanh(mu + std*noise); out[n_sgen] += logp terms
__global__ void __launch_bounds__(256)
k_head(const float* __restrict__ h, const int* __restrict__ map,
       const float* __restrict__ noise,
       const float* __restrict__ Wmu, const float* __restrict__ bmu,
       const float* __restrict__ Wls, const float* __restrict__ bls,
       float* __restrict__ out, int n_sgen) {
  __shared__ float red[256];
  const int i = blockIdx.x * 256 + threadIdx.x;
  float partial = 0.0f;
  if (i < n_sgen) {
    const float* z = h + (size_t)map[i] * HID;
    float mu = 0.0f, ls = 0.0f;
#pragma unroll 4
    for (int c = 0; c < HID; c += 4) {
      float4 zv = *(const float4*)(z + c);
      zv.x = fmaxf(zv.x, 0.0f); zv.y = fmaxf(zv.y, 0.0f);
      zv.z = fmaxf(zv.z, 0.0f); zv.w = fmaxf(zv.w, 0.0f);
      float4 wm = *(const float4*)(Wmu + c);
      float4 wl = *(const float4*)(Wls + c);
      mu += zv.x * wm.x + zv.y * wm.y + zv.z * wm.z + zv.w * wm.w;
      ls += zv.x * wl.x + zv.y * wl.y + zv.z * wl.z + zv.w * wl.w;
    }
    mu += bmu[0];
    ls += bls[0];
    ls = fminf(fmaxf(ls, -20.0f), 2.0f);
    const float sd = expf(ls);
    const float nz = noise[i];
    const float a = tanhf(mu + sd * nz);
    out[i] = a;
    partial = (-0.5f * nz * nz - ls - 0.5f * LOG_2PI)
              - logf(1.0f - a * a + 1e-6f);
  }
  red[threadIdx.x] = partial;
  __syncthreads();
  for (int s = 128; s > 0; s >>= 1) {
    if (threadIdx.x < s) red[threadIdx.x] += red[threadIdx.x + s];
    __syncthreads();
  }
  if (threadIdx.x == 0) atomicAdd(out + n_sgen, red[0]);
}

// ---------------- launcher ----------------

extern "C" void kernel_launch(void* const* d_in, const int* in_sizes, int n_in,
                              void* d_out, int out_size, void* d_ws, size_t ws_size,
                              hipStream_t stream) {
  const float* x     = (const float*)d_in[0];
  const int*   ei    = (const int*)d_in[1];
  const int*   sgen  = (const int*)d_in[2];
  const float* noise = (const float*)d_in[3];
  const float* W_in  = (const float*)d_in[4];
  const float* b_in  = (const float*)d_in[5];
  const float* W1    = (const float*)d_in[6];
  const float* b1    = (const float*)d_in[7];
  const float* W2    = (const float*)d_in[8];
  const float* b2    = (const float*)d_in[9];
  const float* W_mu  = (const float*)d_in[10];
  const float* b_mu  = (const float*)d_in[11];
  const float* W_ls  = (const float*)d_in[12];
  const float* b_ls  = (const float*)d_in[13];

  const int n_nodes = in_sizes[0] / 32;
  const int n_edges = in_sizes[1] / 2;
  const int n_sgen  = in_sizes[2];
  const int K_in    = in_sizes[4] / HID;  // 32

  const int* row  = ei;
  const int* colp = ei + n_edges;

  float* ws  = (float*)d_ws;
  float* deg = ws;                          // n_nodes floats; becomes isd in place
  const size_t PAD = 1u << 17;              // 128K-float pad
  const size_t HN = (size_t)n_nodes * HID;
  float* B0 = ws + PAD;
  float* B1 = B0 + HN;
  float* B2 = B1 + HN;
  float* out = (float*)d_out;

  const int T = 256;
  const int mtiles = (n_nodes + 15) >> 4;
  const int gblocks = (mtiles + 7) >> 3;
  const int aggblocks = (n_nodes * 32 + T - 1) / T;
  const unsigned scatblocks = (unsigned)(((long long)n_edges * 32 + T - 1) / T);

  // degree -> inv_sqrt_deg (in place)
  k_zero<<<(n_nodes + T - 1) / T, T, 0, stream>>>(deg, n_nodes);
  k_zero<<<1, 1, 0, stream>>>(out + n_sgen, 1);  // logp accumulator
  k_deg<<<(n_edges + T - 1) / T, T, 0, stream>>>(colp, deg, n_edges);
  k_isd<<<(n_nodes + T - 1) / T, T, 0, stream>>>(deg, n_nodes);

  // h0 = x @ W_in + b_in
  k_gemm_wmma<<<gblocks, T, 0, stream>>>(x, W_in, b_in, B0, n_nodes, K_in, 0);

  // layer 1: hw = h0 @ W1; agg = self-loop + scatter  (B2 holds conv1 pre-relu)
  k_gemm_wmma<<<gblocks, T, 0, stream>>>(B0, W1, nullptr, B1, n_nodes, HID, 0);
  k_agginit<<<aggblocks, T, 0, stream>>>(B1, deg, b1, B2, n_nodes);
  k_scatter<<<scatblocks, T, 0, stream>>>(B1, row, colp, deg, B2, n_edges);

  // layer 2: hw = relu(conv1) @ W2 (relu fused into GEMM A-load); agg into B0
  k_gemm_wmma<<<gblocks, T, 0, stream>>>(B2, W2, nullptr, B1, n_nodes, HID, 1);
  k_agginit<<<aggblocks, T, 0, stream>>>(B1, deg, b2, B0, n_nodes);
  k_scatter<<<scatblocks, T, 0, stream>>>(B1, row, colp, deg, B0, n_edges);

  // head: relu fused into gather + dual 128-dots + tanh rsample + logp reduction
  k_head<<<(n_sgen + T - 1) / T, T, 0, stream>>>(B0, sgen, noise,
                                                 W_mu, b_mu, W_ls, b_ls,
                                                 out, n_sgen);
}